// Dynamical_F_Attn_Cross_Merged_6485400617079
// MI455X (gfx1250) — compile-verified
//
#include <hip/hip_runtime.h>

typedef __attribute__((ext_vector_type(2))) float v2f;
typedef __attribute__((ext_vector_type(8))) float v8f;

#define WAVES 4
#define ROWS_PER_WAVE 16
#define ROWS_PER_BLOCK (WAVES * ROWS_PER_WAVE)

struct Params {
  const float* x;
  const float* w1; const float* b1;
  const float* w2; const float* b2;
  const float* w3; const float* b3;
  const float* Win[4]; const float* bin[4];
  const float* Wo[4];  const float* bo[4];
  const float* Wf; const float* bf;
  float* out;
};

__device__ __forceinline__ v8f wmma4(v2f a, v2f b, v8f c) {
  // V_WMMA_F32_16X16X4_F32 : D(16x16 f32) = A(16x4) * B(4x16) + C
  return __builtin_amdgcn_wmma_f32_16x16x4_f32(false, a, false, b, (short)0, c,
                                               false, false);
}

__device__ __forceinline__ v2f ld2(const float* p) { return *(const v2f*)p; }

__device__ __forceinline__ v8f splat8(float b) {
  v8f r = {b, b, b, b, b, b, b, b};
  return r;
}

__global__ __launch_bounds__(WAVES * 32) void fused_kernel(Params P) {
  extern __shared__ float smem[];
  const int lane = threadIdx.x & 31;
  const int wave = threadIdx.x >> 5;
  const int half = lane >> 4;   // selects K-pair (A/B frags) / M+8 (C frags)
  const int l16  = lane & 15;

  // Per-wave private LDS regions (no cross-wave sharing -> no barriers; LDS is
  // in-order within a wave).
  float* tokens  = smem + wave * (ROWS_PER_WAVE * 8 * 64);           // 8192 f
  float* scratch = smem + WAVES * 8192 + wave * 4096;                // 4096 f
  float* xs      = smem + WAVES * (8192 + 4096) + wave * 128;        // 128 f
  float* oacc    = smem + WAVES * (8192 + 4096) + WAVES * 128 + wave * 16;

  const int rowbase = (blockIdx.x * WAVES + wave) * ROWS_PER_WAVE;

  // Stage the 16x8 x-tile (contiguous, coalesced).
  #pragma unroll
  for (int j = 0; j < 4; ++j) {
    int idx = lane + 32 * j;
    xs[idx] = P.x[rowbase * 8 + idx];
  }
  if (lane < 16) oacc[lane] = P.bf[0];

  float* h1 = scratch;          // 16x128
  float* h2 = scratch + 2048;   // 16x128

  // ---------------- token MLP: per feature d ----------------
  for (int d = 0; d < 8; ++d) {
    // h1 = sin(x[:,d] (x) w1[d] + b1[d])  : 16x128, 64 values per lane
    for (int j = 0; j < 64; ++j) {
      int idx = lane + 32 * j;
      int m = idx >> 7, k = idx & 127;
      h1[idx] = __sinf(xs[m * 8 + d] * P.w1[d * 128 + k] + P.b1[d * 128 + k]);
    }
    // h2 = sin(h1 @ w2[d]^T + b2[d]) : M=16, N=128 (8 tiles), K=128
    // Two N-tiles per pass: one A-frag load feeds two independent WMMA chains.
    for (int np = 0; np < 4; ++np) {
      int nt0 = 2 * np;
      v8f acc0 = splat8(P.b2[d * 128 + nt0 * 16 + l16]);
      v8f acc1 = splat8(P.b2[d * 128 + nt0 * 16 + 16 + l16]);
      const float* wB0 = P.w2 + ((size_t)d * 128 + nt0 * 16 + l16) * 128 + 2 * half;
      const float* wB1 = wB0 + 16 * 128;
      const float* hA  = h1 + l16 * 128 + 2 * half;
      #pragma unroll 8
      for (int kb = 0; kb < 128; kb += 4) {
        v2f a = ld2(hA + kb);
        acc0 = wmma4(a, ld2(wB0 + kb), acc0);
        acc1 = wmma4(a, ld2(wB1 + kb), acc1);
      }
      #pragma unroll
      for (int v = 0; v < 8; ++v) {
        h2[(v + 8 * half) * 128 + nt0 * 16 + l16]      = __sinf(acc0[v]);
        h2[(v + 8 * half) * 128 + nt0 * 16 + 16 + l16] = __sinf(acc1[v]);
      }
    }
    // tokens[:,d,:] = h2 @ w3[d]^T + b3[d] : M=16, N=64 (4 tiles), K=128
    for (int np = 0; np < 2; ++np) {
      int nt0 = 2 * np;
      v8f acc0 = splat8(P.b3[d * 64 + nt0 * 16 + l16]);
      v8f acc1 = splat8(P.b3[d * 64 + nt0 * 16 + 16 + l16]);
      const float* wB0 = P.w3 + ((size_t)d * 64 + nt0 * 16 + l16) * 128 + 2 * half;
      const float* wB1 = wB0 + 16 * 128;
      const float* hA  = h2 + l16 * 128 + 2 * half;
      #pragma unroll 8
      for (int kb = 0; kb < 128; kb += 4) {
        v2f a = ld2(hA + kb);
        acc0 = wmma4(a, ld2(wB0 + kb), acc0);
        acc1 = wmma4(a, ld2(wB1 + kb), acc1);
      }
      #pragma unroll
      for (int v = 0; v < 8; ++v) {
        tokens[((v + 8 * half) * 8 + d) * 64 + nt0 * 16 + l16]      = acc0[v];
        tokens[((v + 8 * half) * 8 + d) * 64 + nt0 * 16 + 16 + l16] = acc1[v];
      }
    }
  }

  // ---------------- 4 MHA blocks: sa, ca, sb, cb ----------------
  float* Qb = scratch;           // 16x64 each (h1/h2 dead now)
  float* Kb = scratch + 1024;
  float* Vb = scratch + 2048;
  float* Ob = scratch + 3072;

  for (int blk = 0; blk < 4; ++blk) {
    const float* Win = P.Win[blk];
    const float* bin = P.bin[blk];
    const float* Wo  = P.Wo[blk];
    const float* bo  = P.bo[blk];
    const int qb  = (blk >= 2) ? 4 : 0;                  // A tokens=0..3, Bt=4..7
    const int kvb = (blk == 1 || blk == 2) ? 4 : 0;
    const float* wfp = P.Wf + (blk >> 1) * 64;

    for (int mt = 0; mt < 4; ++mt) {  // 4 batches per M-tile; 16 batches/wave
      // Q/K/V projections: M=16 (4 batches x seq 4), N=64, K=64
      for (int r = 0; r < 3; ++r) {
        int tb = (r == 0) ? qb : kvb;
        // row m = bsub*4+seq -> tokens[(mt*4+bsub)*8 + tb+seq][k]
        const float* tokA =
            tokens + ((mt * 4 + (l16 >> 2)) * 8 + tb + (l16 & 3)) * 64 + 2 * half;
        float* dst = scratch + r * 1024;
        for (int np = 0; np < 2; ++np) {
          int nt0 = 2 * np;
          v8f acc0 = splat8(bin[r * 64 + nt0 * 16 + l16]);
          v8f acc1 = splat8(bin[r * 64 + nt0 * 16 + 16 + l16]);
          const float* wB0 = Win + (r * 64 + nt0 * 16 + l16) * 64 + 2 * half;
          const float* wB1 = wB0 + 16 * 64;
          #pragma unroll 8
          for (int kb = 0; kb < 64; kb += 4) {
            v2f a = ld2(tokA + kb);
            acc0 = wmma4(a, ld2(wB0 + kb), acc0);
            acc1 = wmma4(a, ld2(wB1 + kb), acc1);
          }
          #pragma unroll
          for (int v = 0; v < 8; ++v) {
            dst[(v + 8 * half) * 64 + nt0 * 16 + l16]      = acc0[v];
            dst[(v + 8 * half) * 64 + nt0 * 16 + 16 + l16] = acc1[v];
          }
        }
      }
      // scores / softmax / A*V : dh=16, each lane handles two (bsub,head,q) rows
      #pragma unroll
      for (int c2 = 0; c2 < 2; ++c2) {
        int c = lane * 2 + c2;
        int bsub = c >> 4, head = (c >> 2) & 3, q = c & 3;
        const float* qrow = Qb + (bsub * 4 + q) * 64 + head * 16;
        float s[4];
        #pragma unroll
        for (int kk = 0; kk < 4; ++kk) {
          const float* krow = Kb + (bsub * 4 + kk) * 64 + head * 16;
          float dot = 0.f;
          #pragma unroll
          for (int t = 0; t < 16; ++t) dot += qrow[t] * krow[t];
          s[kk] = dot * 0.25f;  // 1/sqrt(16)
        }
        float mx = fmaxf(fmaxf(s[0], s[1]), fmaxf(s[2], s[3]));
        float e0 = __expf(s[0] - mx), e1 = __expf(s[1] - mx);
        float e2 = __expf(s[2] - mx), e3 = __expf(s[3] - mx);
        float inv = 1.f / (e0 + e1 + e2 + e3);
        float p0 = e0 * inv, p1 = e1 * inv, p2 = e2 * inv, p3 = e3 * inv;
        const float* v0 = Vb + (bsub * 4 + 0) * 64 + head * 16;
        const float* v1 = Vb + (bsub * 4 + 1) * 64 + head * 16;
        const float* v2r = Vb + (bsub * 4 + 2) * 64 + head * 16;
        const float* v3 = Vb + (bsub * 4 + 3) * 64 + head * 16;
        float* orow = Ob + (bsub * 4 + q) * 64 + head * 16;
        #pragma unroll
        for (int t = 0; t < 16; ++t)
          orow[t] = p0 * v0[t] + p1 * v1[t] + p2 * v2r[t] + p3 * v3[t];
      }
      // Wo projection + pooling: pool = 0.125 * sum over seq (C rows group by 4)
      float c0 = 0.f, c1 = 0.f;
      for (int np = 0; np < 2; ++np) {
        int nt0 = 2 * np;
        v8f acc0 = splat8(bo[nt0 * 16 + l16]);
        v8f acc1 = splat8(bo[nt0 * 16 + 16 + l16]);
        const float* wB0 = Wo + (nt0 * 16 + l16) * 64 + 2 * half;
        const float* wB1 = wB0 + 16 * 64;
        const float* oA  = Ob + l16 * 64 + 2 * half;
        #pragma unroll 8
        for (int kb = 0; kb < 64; kb += 4) {
          v2f a = ld2(oA + kb);
          acc0 = wmma4(a, ld2(wB0 + kb), acc0);
          acc1 = wmma4(a, ld2(wB1 + kb), acc1);
        }
        float wf0 = wfp[nt0 * 16 + l16];
        float wf1 = wfp[nt0 * 16 + 16 + l16];
        // acc[0..3] = rows of bsub {0|2}, acc[4..7] = rows of bsub {1|3}
        c0 += (acc0[0] + acc0[1] + acc0[2] + acc0[3]) * 0.125f * wf0 +
              (acc1[0] + acc1[1] + acc1[2] + acc1[3]) * 0.125f * wf1;
        c1 += (acc0[4] + acc0[5] + acc0[6] + acc0[7]) * 0.125f * wf0 +
              (acc1[4] + acc1[5] + acc1[6] + acc1[7]) * 0.125f * wf1;
      }
      // deterministic butterfly reduction over the 16 lanes of each half-wave
      #pragma unroll
      for (int m = 8; m >= 1; m >>= 1) {
        c0 += __shfl_xor(c0, m, 32);
        c1 += __shfl_xor(c1, m, 32);
      }
      if (l16 == 0) {
        int b0 = mt * 4 + half * 2;
        oacc[b0]     += c0;
        oacc[b0 + 1] += c1;
      }
    }
  }

  // ---------------- final: leaky_relu(acc) ----------------
  if (lane < 16) {
    float v = oacc[lane];
    P.out[rowbase + lane] = (v > 0.f) ? v : 0.01f * v;
  }
}

extern "C" void kernel_launch(void* const* d_in, const int* in_sizes, int n_in,
                              void* d_out, int out_size, void* d_ws, size_t ws_size,
                              hipStream_t stream) {
  Params P;
  P.x  = (const float*)d_in[0];
  P.w1 = (const float*)d_in[1];  P.b1 = (const float*)d_in[2];
  P.w2 = (const float*)d_in[3];  P.b2 = (const float*)d_in[4];
  P.w3 = (const float*)d_in[5];  P.b3 = (const float*)d_in[6];
  // kernel block order: sa, ca, sb, cb  (input order: sa, sb, ca, cb)
  const int base[4] = {7, 15, 11, 19};
  for (int i = 0; i < 4; ++i) {
    P.Win[i] = (const float*)d_in[base[i] + 0];
    P.bin[i] = (const float*)d_in[base[i] + 1];
    P.Wo[i]  = (const float*)d_in[base[i] + 2];
    P.bo[i]  = (const float*)d_in[base[i] + 3];
  }
  P.Wf = (const float*)d_in[23];
  P.bf = (const float*)d_in[24];
  P.out = (float*)d_out;

  const int blocks = out_size / ROWS_PER_BLOCK;  // 65536/64 = 1024
  const size_t smem =
      (size_t)(WAVES * (8192 + 4096 + 128 + 16)) * sizeof(float);  // ~194 KB
  fused_kernel<<<blocks, WAVES * 32, smem, stream>>>(P);
}